// ROIHeads_51625506898097
// MI455X (gfx1250) — compile-verified
//
#include <hip/hip_runtime.h>
#include <hip/hip_bf16.h>
#include <math.h>

#define B_IMG 4
#define N_PROPS 1000
#define NUM_CLASSES 91
#define CM1 90
#define M_CAND (N_PROPS * CM1)      /* 90000 candidates per image      */
#define PADM 131072                 /* next pow2 >= M_CAND             */
#define TOPK 4096
#define DET 100
#define SCORE_THRESH 0.05f
#define NMS_THRESH 0.5f
#define MIN_SIZE 1.0f
#define BBOX_CLIP 4.135166556742356f /* log(1000/16) */

typedef unsigned int u32;
typedef unsigned long long u64;
typedef float v2f __attribute__((ext_vector_type(2)));
typedef float v8f __attribute__((ext_vector_type(8)));

// ---------------------------------------------------------------- init ----
__global__ void k_init(u64* __restrict__ keys, u32* __restrict__ maxbox) {
    int tid = blockIdx.x * blockDim.x + threadIdx.x;
    if (tid < B_IMG) maxbox[tid] = 0u;
    const int npad = PADM - M_CAND;
    if (tid < B_IMG * npad) {
        int img = tid / npad;
        int off = M_CAND + tid - img * npad;
        keys[(size_t)img * PADM + off] = 0ull; // sorts last (descending)
    }
}

// ------------------------------------------- softmax + decode + keys ----
__global__ __launch_bounds__(128) void k_decode(
    const float* __restrict__ cls, const float* __restrict__ bpred,
    const float* __restrict__ props, const int* __restrict__ imsz,
    u64* __restrict__ keys, float4* __restrict__ cboxes,
    u32* __restrict__ maxbox)
{
    int row  = blockIdx.x;                 // [0, B*N)
    int img  = row / N_PROPS;
    int prop = row - img * N_PROPS;
    int t = threadIdx.x;
    __shared__ float red[128];

    float logit = (t < NUM_CLASSES) ? cls[(size_t)row * NUM_CLASSES + t] : -INFINITY;
    red[t] = logit; __syncthreads();
    for (int s = 64; s > 0; s >>= 1) { if (t < s) red[t] = fmaxf(red[t], red[t + s]); __syncthreads(); }
    float mx = red[0]; __syncthreads();
    float e = (t < NUM_CLASSES) ? expf(logit - mx) : 0.f;
    red[t] = e; __syncthreads();
    for (int s = 64; s > 0; s >>= 1) { if (t < s) red[t] += red[t + s]; __syncthreads(); }
    float score = e / red[0];
    __syncthreads();

    float H = (float)imsz[img * 2 + 0];
    float W = (float)imsz[img * 2 + 1];
    float px1 = props[row * 4 + 0], py1 = props[row * 4 + 1];
    float px2 = props[row * 4 + 2], py2 = props[row * 4 + 3];
    float pw = px2 - px1, ph = py2 - py1;
    float pcx = px1 + 0.5f * pw, pcy = py1 + 0.5f * ph;

    float coordmax = 0.f;
    if (t >= 1 && t <= CM1) {
        const float* c = bpred + (size_t)row * (NUM_CLASSES * 4) + t * 4;
        float dx = c[0] / 10.0f, dy = c[1] / 10.0f;
        float dw = fminf(c[2] / 5.0f, BBOX_CLIP), dh = fminf(c[3] / 5.0f, BBOX_CLIP);
        float cx = dx * pw + pcx, cy = dy * ph + pcy;
        float w2 = 0.5f * expf(dw) * pw, h2 = 0.5f * expf(dh) * ph;
        float x1 = fminf(fmaxf(cx - w2, 0.f), W);
        float y1 = fminf(fmaxf(cy - h2, 0.f), H);
        float x2 = fminf(fmaxf(cx + w2, 0.f), W);
        float y2 = fminf(fmaxf(cy + h2, 0.f), H);
        coordmax = fmaxf(fmaxf(x1, y1), fmaxf(x2, y2));
        bool valid = (score > SCORE_THRESH) && ((x2 - x1) >= MIN_SIZE) && ((y2 - y1) >= MIN_SIZE);
        float masked = valid ? score : -INFINITY;
        u32 sb = __float_as_uint(masked);
        u32 ord = (sb & 0x80000000u) ? ~sb : (sb | 0x80000000u);
        u32 idx = (u32)(prop * CM1 + (t - 1));
        keys[(size_t)img * PADM + idx] = ((u64)ord << 32) | (u64)(0xFFFFFFFFu - idx);
        cboxes[(size_t)img * M_CAND + idx] = make_float4(x1, y1, x2, y2);
    }
    red[t] = coordmax; __syncthreads();
    for (int s = 64; s > 0; s >>= 1) { if (t < s) red[t] = fmaxf(red[t], red[t + s]); __syncthreads(); }
    if (t == 0) atomicMax(maxbox + img, __float_as_uint(red[0]));
}

// -------------------------------------------------- bitonic sort (desc) ----
__global__ void k_bitonic_global(u64* __restrict__ keys, int k, int j) {
    int tid = blockIdx.x * blockDim.x + threadIdx.x;
    int i   = tid & (PADM - 1);
    int img = tid >> 17;
    int p   = i ^ j;
    if (p <= i) return;
    u64* a = keys + ((size_t)img << 17);
    u64 x = a[i], y = a[p];
    bool up = ((i & k) == 0);          // descending network
    if (up ? (x < y) : (x > y)) { a[i] = y; a[p] = x; }
}

__global__ __launch_bounds__(1024) void k_bitonic_local_sort(u64* __restrict__ keys) {
    __shared__ u64 sh[2048];
    int t = threadIdx.x;
    size_t base = (size_t)blockIdx.x * 2048;
    int baseInImg = (blockIdx.x & 63) * 2048;
    sh[t] = keys[base + t];
    sh[t + 1024] = keys[base + t + 1024];
    __syncthreads();
    for (int k = 2; k <= 2048; k <<= 1) {
        for (int j = k >> 1; j > 0; j >>= 1) {
            int i = ((t & ~(j - 1)) << 1) | (t & (j - 1));
            int p = i | j;
            u64 x = sh[i], y = sh[p];
            bool up = (((baseInImg + i) & k) == 0);
            if (up ? (x < y) : (x > y)) { sh[i] = y; sh[p] = x; }
            __syncthreads();
        }
    }
    keys[base + t] = sh[t];
    keys[base + t + 1024] = sh[t + 1024];
}

__global__ __launch_bounds__(1024) void k_bitonic_local_merge(u64* __restrict__ keys, int k) {
    __shared__ u64 sh[2048];
    int t = threadIdx.x;
    size_t base = (size_t)blockIdx.x * 2048;
    int baseInImg = (blockIdx.x & 63) * 2048;
    bool up = ((baseInImg & k) == 0);  // uniform per chunk (k > 2048)
    sh[t] = keys[base + t];
    sh[t + 1024] = keys[base + t + 1024];
    __syncthreads();
    for (int j = 1024; j > 0; j >>= 1) {
        int i = ((t & ~(j - 1)) << 1) | (t & (j - 1));
        int p = i | j;
        u64 x = sh[i], y = sh[p];
        if (up ? (x < y) : (x > y)) { sh[i] = y; sh[p] = x; }
        __syncthreads();
    }
    keys[base + t] = sh[t];
    keys[base + t + 1024] = sh[t + 1024];
}

// ------------------------------------------------- gather top-4096 ----
__global__ __launch_bounds__(256) void k_gather(
    const u64* __restrict__ keys, const float4* __restrict__ cboxes,
    const u32* __restrict__ maxbox,
    float4* __restrict__ tkBox, float4* __restrict__ tkNms,
    float* __restrict__ tkScore, int* __restrict__ tkLabel,
    float* __restrict__ tkArea, u32* __restrict__ tkValid)
{
    int tid = blockIdx.x * 256 + threadIdx.x;
    int img = tid >> 12;
    int r   = tid & (TOPK - 1);
    u64 key = keys[((size_t)img << 17) + r];
    u32 hi = (u32)(key >> 32), lo = (u32)key;
    bool valid = hi > 0x007FFFFFu;     // orderable(-inf) == 0x007FFFFF
    float score = 0.f, area = 0.f;
    int label = 0;
    float4 box = make_float4(0.f, 0.f, 0.f, 0.f);
    float4 nb  = make_float4(0.f, 0.f, 0.f, 0.f);
    if (valid) {
        u32 idx = 0xFFFFFFFFu - lo;
        u32 sb  = (hi & 0x80000000u) ? (hi & 0x7FFFFFFFu) : ~hi;
        score = __uint_as_float(sb);
        label = (int)(idx % CM1) + 1;
        box = cboxes[(size_t)img * M_CAND + idx];
        float off = (float)label * (__uint_as_float(maxbox[img]) + 1.0f);
        nb = make_float4(box.x + off, box.y + off, box.z + off, box.w + off);
        area = (box.z - box.x) * (box.w - box.y);
    }
    tkBox[tid] = box; tkNms[tid] = nb;
    tkScore[tid] = score; tkLabel[tid] = label; tkArea[tid] = area;
    u32 bal = (u32)__ballot(valid);
    if ((threadIdx.x & 31) == 0) tkValid[(img << 7) + (r >> 5)] = bal;
}

// ------------------------- IoU suppression mask (WMMA union outer-sum) ----
__global__ __launch_bounds__(256) void k_mask(
    const float4* __restrict__ tkNms, const float* __restrict__ tkArea,
    u32* __restrict__ mask)
{
    __shared__ float4 shBox[512];
    __shared__ float  shArea[512];
    __shared__ u32    shMask[2048];            // [128 block-rows][16 words]
    int img  = blockIdx.y;
    int wave = threadIdx.x >> 5, lane = threadIdx.x & 31;
    int rowBase = blockIdx.x * 128 + wave * 16;
    const float4* nmsI  = tkNms + ((size_t)img << 12);
    const float*  areaI = tkArea + ((size_t)img << 12);
    u32* maskI = mask + ((size_t)img << 19);   // img*4096*128

    int hiOff = (lane >= 16) ? 8 : 0;
    float4 rbx[8];
#pragma unroll
    for (int v = 0; v < 8; ++v) rbx[v] = nmsI[rowBase + v + hiOff];

    // A fragment (16x4 f32): lanes 0-15 hold M=lane, K0=area_row, K1=1; hi lanes K2/K3 = 0
    v2f aF;
    aF.x = (lane < 16) ? areaI[rowBase + lane] : 0.f;
    aF.y = (lane < 16) ? 1.f : 0.f;

    for (int chunk = 0; chunk < TOPK; chunk += 512) {
        __syncthreads();   // protects shBox/shArea/shMask reuse across chunks
        {
            // async global->LDS staging of 512 column boxes + areas (ASYNCcnt path)
            int idx = threadIdx.x;
            unsigned long long g;
            unsigned l;
            g = (unsigned long long)(size_t)(nmsI + chunk + idx);
            l = (unsigned)(size_t)&shBox[idx];
            asm volatile("global_load_async_to_lds_b128 %0, %1, off" :: "v"(l), "v"(g) : "memory");
            g = (unsigned long long)(size_t)(nmsI + chunk + idx + 256);
            l = (unsigned)(size_t)&shBox[idx + 256];
            asm volatile("global_load_async_to_lds_b128 %0, %1, off" :: "v"(l), "v"(g) : "memory");
            g = (unsigned long long)(size_t)(areaI + chunk + idx);
            l = (unsigned)(size_t)&shArea[idx];
            asm volatile("global_load_async_to_lds_b32 %0, %1, off" :: "v"(l), "v"(g) : "memory");
            g = (unsigned long long)(size_t)(areaI + chunk + idx + 256);
            l = (unsigned)(size_t)&shArea[idx + 256];
            asm volatile("global_load_async_to_lds_b32 %0, %1, off" :: "v"(l), "v"(g) : "memory");
            asm volatile("s_wait_asynccnt 0" ::: "memory");
        }
        __syncthreads();
        for (int cw = 0; cw < 16; ++cw) {
            int lbase = cw * 32;
            int c0 = chunk + lbase;
            int l16 = lane & 15;
            float4 cbA = shBox[lbase + l16];
            float4 cbB = shBox[lbase + 16 + l16];
            float  caA = shArea[lbase + l16];        // unconditional (no exec juggling)
            float  caB = shArea[lbase + 16 + l16];
            // B fragment (4x16 f32): lanes 0-15 hold N=lane, K0=1, K1=area_col; hi lanes 0
            v2f bFA, bFB;
            bFA.x = (lane < 16) ? 1.f : 0.f;
            bFA.y = (lane < 16) ? caA : 0.f;
            bFB.x = bFA.x;
            bFB.y = (lane < 16) ? caB : 0.f;
            v8f cz = {0.f, 0.f, 0.f, 0.f, 0.f, 0.f, 0.f, 0.f};
            // D[m][n] = area_row[m] + area_col[n]  (exact f32, co-executes with VALU)
            v8f SA = __builtin_amdgcn_wmma_f32_16x16x4_f32(false, aF, false, bFA, (short)0, cz, false, false);
            v8f SB = __builtin_amdgcn_wmma_f32_16x16x4_f32(false, aF, false, bFB, (short)0, cz, false, false);
            int colA = c0 + l16, colB = colA + 16;
#pragma unroll
            for (int v = 0; v < 8; ++v) {
                int rowI = rowBase + v + hiOff;
                float4 rb = rbx[v];
                float iwA = fminf(rb.z, cbA.z) - fmaxf(rb.x, cbA.x);
                float ihA = fminf(rb.w, cbA.w) - fmaxf(rb.y, cbA.y);
                float inA = fmaxf(iwA, 0.f) * fmaxf(ihA, 0.f);
                bool sA = (colA > rowI) && (inA > NMS_THRESH * (SA[v] - inA));
                float iwB = fminf(rb.z, cbB.z) - fmaxf(rb.x, cbB.x);
                float ihB = fminf(rb.w, cbB.w) - fmaxf(rb.y, cbB.y);
                float inB = fmaxf(iwB, 0.f) * fmaxf(ihB, 0.f);
                bool sB = (colB > rowI) && (inB > NMS_THRESH * (SB[v] - inB));
                u32 bA = (u32)__ballot(sA);
                u32 bB = (u32)__ballot(sB);
                if (lane == 0) {
                    u32 wlo = (bA & 0xFFFFu) | (bB << 16);          // block-row wave*16+v
                    u32 whi = (bA >> 16) | (bB & 0xFFFF0000u);      // block-row wave*16+v+8
                    shMask[(wave * 16 + v) * 16 + cw] = wlo;
                    shMask[(wave * 16 + v + 8) * 16 + cw] = whi;
                }
            }
        }
        __syncthreads();
        // coalesced b128 flush of the 128x16 mask tile (8 KB) for this chunk
        {
            const uint4* sm = (const uint4*)shMask;
            int wordbase = chunk >> 5;                 // 16-word aligned
            for (int q = threadIdx.x; q < 512; q += 256) {
                int r = q >> 2, sub = q & 3;
                uint4 vm = sm[q];
                *(uint4*)(maskI + (size_t)(blockIdx.x * 128 + r) * 128 + wordbase + sub * 4) = vm;
            }
        }
    }
}

// --------------------------------------------- serial NMS scan (1 wave) ----
__global__ __launch_bounds__(32) void k_nms_serial(
    const u32* __restrict__ tkValid, const u32* __restrict__ mask,
    u32* __restrict__ keepfin)
{
    __shared__ u32 remS[128];
    __shared__ u32 keepS[128];
    int img = blockIdx.x;
    int lane = threadIdx.x;
    volatile u32* rem = remS;
    volatile u32* kp  = keepS;
    for (int w = lane; w < 128; w += 32) { remS[w] = 0u; keepS[w] = tkValid[(img << 7) + w]; }
    __syncthreads();
    const u32* maskI = mask + ((size_t)img << 19);
    for (int i = 0; i < TOPK; ++i) {
        u32 kw = kp[i >> 5], rw = rem[i >> 5];
        if (lane == 0 && i + 1 < TOPK) __builtin_prefetch(maskI + (size_t)(i + 1) * 128, 0, 1);
        if (((kw >> (i & 31)) & 1u) && !((rw >> (i & 31)) & 1u)) {
            const u32* mrow = maskI + (size_t)i * 128;
            for (int w = lane; w < 128; w += 32) rem[w] |= mrow[w];
        }
    }
    __syncthreads();
    for (int w = lane; w < 128; w += 32) keepfin[(img << 7) + w] = keepS[w] & ~remS[w];
}

// ---------------------------------- final top-100 (order-preserving) ----
__global__ __launch_bounds__(256) void k_finalize(
    const u32* __restrict__ keepfin, const float4* __restrict__ tkBox,
    const float* __restrict__ tkScore, const int* __restrict__ tkLabel,
    float* __restrict__ outBoxes, float* __restrict__ outScores,
    int* __restrict__ outLabels)
{
    int img = blockIdx.x;
    int t = threadIdx.x;
    __shared__ u32 sc[256];
    u32 word = keepfin[(img << 7) + (t >> 1)];
    u32 bits = ((t & 1) ? (word >> 16) : word) & 0xFFFFu;
    u32 cnt = (u32)__popc(bits);
    sc[t] = cnt;
    __syncthreads();
    for (int s = 1; s < 256; s <<= 1) {
        u32 v = (t >= s) ? sc[t - s] : 0u;
        __syncthreads();
        sc[t] += v;
        __syncthreads();
    }
    u32 total = sc[255];
    u32 rank = sc[t] - cnt;            // exclusive prefix
    // init only slots that will NOT be written by the scatter (race-free)
    for (int r = t; r < DET; r += 256) {
        if ((u32)r >= total) {
            outBoxes[(img * DET + r) * 4 + 0] = 0.f;
            outBoxes[(img * DET + r) * 4 + 1] = 0.f;
            outBoxes[(img * DET + r) * 4 + 2] = 0.f;
            outBoxes[(img * DET + r) * 4 + 3] = 0.f;
            outScores[img * DET + r] = 0.f;
            outLabels[img * DET + r] = -1;
        }
    }
    int baseI = (img << 12) + t * 16;
    for (int e = 0; e < 16; ++e) {
        if ((bits >> e) & 1u) {
            if (rank < DET) {
                int i = baseI + e;
                float4 bx = tkBox[i];
                outBoxes[(img * DET + rank) * 4 + 0] = bx.x;
                outBoxes[(img * DET + rank) * 4 + 1] = bx.y;
                outBoxes[(img * DET + rank) * 4 + 2] = bx.z;
                outBoxes[(img * DET + rank) * 4 + 3] = bx.w;
                outScores[img * DET + rank] = tkScore[i];
                outLabels[img * DET + rank] = tkLabel[i];
            }
            rank++;
        }
    }
}

// --------------------------------------------------------------- launch ----
extern "C" void kernel_launch(void* const* d_in, const int* in_sizes, int n_in,
                              void* d_out, int out_size, void* d_ws, size_t ws_size,
                              hipStream_t stream) {
    (void)in_sizes; (void)n_in; (void)out_size; (void)ws_size;
    const float* cls   = (const float*)d_in[0];
    const float* bpred = (const float*)d_in[1];
    const float* props = (const float*)d_in[2];
    const int*   imsz  = (const int*)d_in[3];

    char* ws = (char*)d_ws;
    size_t off = 0;
    auto alloc = [&](size_t bytes) {
        size_t o = off;
        off = (off + bytes + 255) & ~(size_t)255;
        return o;
    };
    u64*    keys    = (u64*)   (ws + alloc(sizeof(u64) * B_IMG * PADM));              // 4 MB
    float4* cboxes  = (float4*)(ws + alloc(sizeof(float4) * (size_t)B_IMG * M_CAND)); // 5.8 MB
    u32*    maxbox  = (u32*)   (ws + alloc(sizeof(u32) * B_IMG));
    float4* tkBox   = (float4*)(ws + alloc(sizeof(float4) * B_IMG * TOPK));
    float4* tkNms   = (float4*)(ws + alloc(sizeof(float4) * B_IMG * TOPK));
    float*  tkScore = (float*) (ws + alloc(sizeof(float) * B_IMG * TOPK));
    int*    tkLabel = (int*)   (ws + alloc(sizeof(int) * B_IMG * TOPK));
    float*  tkArea  = (float*) (ws + alloc(sizeof(float) * B_IMG * TOPK));
    u32*    tkValid = (u32*)   (ws + alloc(sizeof(u32) * B_IMG * (TOPK / 32)));
    u32*    keepfin = (u32*)   (ws + alloc(sizeof(u32) * B_IMG * (TOPK / 32)));
    u32*    maskbuf = (u32*)   (ws + alloc(sizeof(u32) * (size_t)B_IMG * TOPK * (TOPK / 32))); // 8 MB

    k_init<<<(B_IMG * (PADM - M_CAND) + 255) / 256, 256, 0, stream>>>(keys, maxbox);
    k_decode<<<B_IMG * N_PROPS, 128, 0, stream>>>(cls, bpred, props, imsz, keys, cboxes, maxbox);

    k_bitonic_local_sort<<<B_IMG * (PADM / 2048), 1024, 0, stream>>>(keys);
    for (int k = 4096; k <= PADM; k <<= 1) {
        for (int j = k >> 1; j >= 2048; j >>= 1)
            k_bitonic_global<<<(B_IMG * PADM) / 256, 256, 0, stream>>>(keys, k, j);
        k_bitonic_local_merge<<<B_IMG * (PADM / 2048), 1024, 0, stream>>>(keys, k);
    }

    k_gather<<<(B_IMG * TOPK) / 256, 256, 0, stream>>>(keys, cboxes, maxbox,
        tkBox, tkNms, tkScore, tkLabel, tkArea, tkValid);

    dim3 gE(TOPK / 128, B_IMG);
    k_mask<<<gE, 256, 0, stream>>>(tkNms, tkArea, maskbuf);

    k_nms_serial<<<B_IMG, 32, 0, stream>>>(tkValid, maskbuf, keepfin);

    float* outBoxes  = (float*)d_out;
    float* outScores = outBoxes + B_IMG * DET * 4;
    int*   outLabels = (int*)(outScores + B_IMG * DET);
    k_finalize<<<B_IMG, 256, 0, stream>>>(keepfin, tkBox, tkScore, tkLabel,
                                          outBoxes, outScores, outLabels);
}